// LatentRayMarcher_30322469109940
// MI455X (gfx1250) — compile-verified
//
#include <hip/hip_runtime.h>
#include <hip/hip_bf16.h>
#include <stdint.h>

#define MAX_STEPS     64
#define BASE_EPS      0.01f
#define LATENT_SCALE  0.18215f
#define TAN_HALF_FOV  0.5773502691896258f   // tan(30 deg)
#define NORMAL_EPS    0.01f

typedef unsigned int u32x4 __attribute__((ext_vector_type(4)));
typedef int          i32x4 __attribute__((ext_vector_type(4)));
typedef int          i32x8 __attribute__((ext_vector_type(8)));

// ---- order-preserving float <-> uint keys for atomic min/max ----
__device__ __forceinline__ unsigned fkey(float f) {
    unsigned b = __float_as_uint(f);
    return (b & 0x80000000u) ? ~b : (b | 0x80000000u);
}
__device__ __forceinline__ float unfkey(unsigned k) {
    unsigned b = (k & 0x80000000u) ? (k & 0x7FFFFFFFu) : ~k;
    return __uint_as_float(b);
}
__device__ __forceinline__ unsigned umin2(unsigned a, unsigned b) { return a < b ? a : b; }
__device__ __forceinline__ unsigned umax2(unsigned a, unsigned b) { return a > b ? a : b; }

// ---- bilinear sampling of a 64x64 channel, align_corners=True, border pad ----
struct Bil { int i00, i01, i10, i11; float wx, wy; };

__device__ __forceinline__ Bil bil_setup(float x, float y) {
    float ix = fminf(fmaxf((x + 1.0f) * 0.5f * 63.0f, 0.0f), 63.0f);
    float iy = fminf(fmaxf((y + 1.0f) * 0.5f * 63.0f, 0.0f), 63.0f);
    float fx0 = floorf(ix), fy0 = floorf(iy);
    int x0 = (int)fx0, y0 = (int)fy0;
    int x1 = min(x0 + 1, 63), y1 = min(y0 + 1, 63);
    Bil b;
    b.wx = ix - fx0; b.wy = iy - fy0;
    b.i00 = y0 * 64 + x0; b.i01 = y0 * 64 + x1;
    b.i10 = y1 * 64 + x0; b.i11 = y1 * 64 + x1;
    return b;
}

__device__ __forceinline__ float bil_fetch(const float* ch, const Bil& b) {
    float v00 = ch[b.i00], v01 = ch[b.i01], v10 = ch[b.i10], v11 = ch[b.i11];
    float top = v00 * (1.0f - b.wx) + v01 * b.wx;
    float bot = v10 * (1.0f - b.wx) + v11 * b.wx;
    return top * (1.0f - b.wy) + bot * b.wy;
}

// ---- main ray-march kernel (first in file so the TDM issue shows in disasm) ----
__global__ __launch_bounds__(256) void lrm_march(
    const float* __restrict__ latent,   // (1,4,64,64) f32, contiguous
    const int*   __restrict__ pw,
    const int*   __restrict__ ph,
    float*       __restrict__ out,
    int N,
    unsigned*    __restrict__ ws)
{
    __shared__ float s_lat[4 * 64 * 64];  // 64 KB: whole latent, [c][y][x]

    // Wave 0 issues a single TDM DMA: 16384 f32 elements global -> LDS.
    if (threadIdx.x < 32) {
        unsigned lds_off = (unsigned)(uintptr_t)(void*)&s_lat[0];
        unsigned long long ga = (unsigned long long)(uintptr_t)latent;
        u32x4 g0;
        g0[0] = 1u;                                   // count=1, user descriptor
        g0[1] = lds_off;                              // lds_addr (bytes)
        g0[2] = (unsigned)(ga & 0xFFFFFFFFull);       // global_addr[31:0]
        g0[3] = ((unsigned)(ga >> 32) & 0x01FFFFFFu)  // global_addr[56:32]
                | 0x80000000u;                        // type=2 ("image")
        i32x8 g1;
        g1[0] = (int)(2u << 16);       // data_size=2 -> 4 bytes; wg_mask=0 (not in cluster)
        g1[1] = (int)(16384u << 16);   // tensor_dim0[15:0]=16384 in bits[31:16]
        g1[2] = (int)(1u << 16);       // tensor_dim0[31:16]=0 ; tensor_dim1=1
        g1[3] = (int)(16384u << 16);   // tensor_dim1 hi=0 ; tile_dim0=16384
        g1[4] = 1;                     // tile_dim1=1, tile_dim2=0
        g1[5] = 16384;                 // tensor_dim0_stride[31:0]
        g1[6] = 0;                     // stride0 hi / stride1 lo
        g1[7] = 0;                     // stride1 hi
        i32x4 z4 = {0, 0, 0, 0};
#if defined(__clang_major__) && (__clang_major__ >= 23)
        i32x8 z8 = {0, 0, 0, 0, 0, 0, 0, 0};
        __builtin_amdgcn_tensor_load_to_lds(g0, g1, z4, z4, z8, 0);
#else
        __builtin_amdgcn_tensor_load_to_lds(g0, g1, z4, z4, 0);
#endif
        __builtin_amdgcn_s_wait_tensorcnt(0);
    }
    __syncthreads();

    int idx = blockIdx.x * blockDim.x + threadIdx.x;
    unsigned kmin = 0xFFFFFFFFu, kmax = 0u;

    if (idx < N) {
        int W = pw[0], H = ph[0];
        int pxi = idx % W, pyi = idx / W;
        float aspect = (float)W / (float)H;
        float u = (2.0f * (float)pxi / (float)(W - 1) - 1.0f) * aspect * TAN_HALF_FOV;
        float v = (2.0f * (float)pyi / (float)(H - 1) - 1.0f) * TAN_HALF_FOV;
        float dx = u, dy = -v, dz = 1.0f;
        float il = 1.0f / fmaxf(sqrtf(dx * dx + dy * dy + dz * dz), 1e-12f);
        dx *= il; dy *= il; dz *= il;

        float px = 0.0f, py = 0.0f, pz = -2.0f, dist = 0.0f;
        bool hit = false;
        // Reference scans all 64 steps (position keeps advancing after hit):
        // no early exit allowed.
        for (int i = 0; i < MAX_STEPS; ++i) {
            Bil b = bil_setup(px, py);
            float s = bil_fetch(s_lat, b) * LATENT_SCALE;  // SDF = ch0 sample
            hit = hit || (fabsf(s) < BASE_EPS);
            float adv = fmaxf(s, BASE_EPS);
            px += dx * adv; py += dy * adv; pz += dz * adv;
            dist += fabsf(s);
        }

        // Finite-difference normal (z taps cancel exactly: SDF ignores z)
        float sxp = bil_fetch(s_lat, bil_setup(px + NORMAL_EPS, py));
        float sxm = bil_fetch(s_lat, bil_setup(px - NORMAL_EPS, py));
        float syp = bil_fetch(s_lat, bil_setup(px, py + NORMAL_EPS));
        float sym = bil_fetch(s_lat, bil_setup(px, py - NORMAL_EPS));
        float nx = (sxp - sxm) * LATENT_SCALE;
        float ny = (syp - sym) * LATENT_SCALE;
        float nil = 1.0f / fmaxf(sqrtf(nx * nx + ny * ny), 1e-12f);
        float n0 = nx * nil, n1 = ny * nil, n2 = 0.0f;

        // Color sample: channels 1..3 at final position
        Bil bc = bil_setup(px, py);
        float c1 = bil_fetch(s_lat + 4096,  bc) * LATENT_SCALE;
        float c2 = bil_fetch(s_lat + 8192,  bc) * LATENT_SCALE;
        float c3 = bil_fetch(s_lat + 12288, bc) * LATENT_SCALE;

        float hf = hit ? 1.0f : 0.0f;
        float d  = fminf(fmaxf((dist - 0.1f) / 9.9f, 0.0f), 1.0f);
        float depth = (1.0f - d) * hf;

        size_t NN = (size_t)N;
        out[idx] = depth;                               // [0, N)   depth
        float* pN = out + NN;                           // [N, 4N)  normal
        pN[3 * idx + 0] = n0; pN[3 * idx + 1] = n1; pN[3 * idx + 2] = n2;
        float* pNI = out + 4 * NN;                      // [4N,7N)  normal_image
        pNI[3 * idx + 0] = (n0 * 0.5f + 0.5f) * hf;
        pNI[3 * idx + 1] = (n1 * 0.5f + 0.5f) * hf;
        pNI[3 * idx + 2] = (n2 * 0.5f + 0.5f) * hf;
        out[7 * NN + idx] = hf;                         // [7N,8N)  hit
        float* pC = out + 8 * NN;                       // [8N,11N) color (raw, pass 2 normalizes)
        pC[3 * idx + 0] = c1; pC[3 * idx + 1] = c2; pC[3 * idx + 2] = c3;
        out[11 * NN + idx] = dist;                      // [11N,12N) distance
        float* pP = out + 12 * NN;                      // [12N,15N) position
        pP[3 * idx + 0] = px; pP[3 * idx + 1] = py; pP[3 * idx + 2] = pz;

        kmin = umin2(umin2(fkey(c1), fkey(c2)), fkey(c3));
        kmax = umax2(umax2(fkey(c1), fkey(c2)), fkey(c3));
    }

    // wave32 butterfly reduction, then one atomic per wave
    for (int off = 16; off > 0; off >>= 1) {
        unsigned om = (unsigned)__shfl_xor((int)kmin, off, 32);
        unsigned oM = (unsigned)__shfl_xor((int)kmax, off, 32);
        kmin = umin2(kmin, om);
        kmax = umax2(kmax, oM);
    }
    if ((threadIdx.x & 31u) == 0u) {
        atomicMin(&ws[0], kmin);
        atomicMax(&ws[1], kmax);
    }
}

// ---- workspace init (runs every call; capture-safe) ----
__global__ void lrm_ws_init(unsigned* ws) {
    ws[0] = 0xFFFFFFFFu;  // min key
    ws[1] = 0x00000000u;  // max key
}

// ---- pass 2: normalize color in place using global min/max ----
__global__ __launch_bounds__(256) void lrm_color_norm(
    float* __restrict__ out, const unsigned* __restrict__ ws, int N)
{
    int i = blockIdx.x * blockDim.x + threadIdx.x;
    int n3 = 3 * N;
    if (i >= n3) return;
    float mn = unfkey(ws[0]);
    float mx = unfkey(ws[1]);
    float inv = 1.0f / (mx - mn + 1e-8f);
    float* color = out + (size_t)8 * (size_t)N;
    color[i] = (color[i] - mn) * inv;
}

extern "C" void kernel_launch(void* const* d_in, const int* in_sizes, int n_in,
                              void* d_out, int out_size, void* d_ws, size_t ws_size,
                              hipStream_t stream) {
    const float* latent = (const float*)d_in[0];
    const int*   pw     = (const int*)d_in[1];
    const int*   ph     = (const int*)d_in[2];
    float*       out    = (float*)d_out;
    unsigned*    ws     = (unsigned*)d_ws;

    int N = out_size / 15;  // depth+3*normal+3*nimg+hit+3*color+dist+3*pos

    lrm_ws_init<<<1, 1, 0, stream>>>(ws);
    lrm_march<<<(N + 255) / 256, 256, 0, stream>>>(latent, pw, ph, out, N, ws);
    lrm_color_norm<<<(3 * N + 255) / 256, 256, 0, stream>>>(out, ws, N);
}